// HyperNet_68513318306030
// MI455X (gfx1250) — compile-verified
//
#include <hip/hip_runtime.h>

#define B_ROWS   262144
#define EMB      16
#define DICT     100
#define HID      32
#define POUT     512
#define NTILES   (B_ROWS / 16)
#define TPB      256
#define NBLOCKS  512
#define WPB      (TPB / 32)
#define TOTAL_WAVES (NBLOCKS * WPB)

typedef float v2f __attribute__((ext_vector_type(2)));
typedef float v8f __attribute__((ext_vector_type(8)));

__device__ __forceinline__ float dot4(float4 a, float4 b) {
    return fmaf(a.x, b.x, fmaf(a.y, b.y, fmaf(a.z, b.z, a.w * b.w)));
}
__device__ __forceinline__ float dsq4(float4 a, float4 b) {
    float x = a.x - b.x, y = a.y - b.y, z = a.z - b.z, w = a.w - b.w;
    return fmaf(x, x, fmaf(y, y, fmaf(z, z, w * w)));
}
__device__ __forceinline__ float lrelu(float s) { return s > 0.0f ? s : 0.01f * s; }

__global__ void init_tail_kernel(const float* __restrict__ usage_in,
                                 float* __restrict__ tail) {
    int t = threadIdx.x;
    if (t == 0) tail[0] = 0.0f;               // loss
    if (t < DICT) tail[1 + t] = usage_in[t];  // usage_new starts at usage
}

__global__ __launch_bounds__(TPB) void vqvae_hyper_kernel(
    const float* __restrict__ embed, const float* __restrict__ codebook,
    const float* __restrict__ W1, const float* __restrict__ b1,
    const float* __restrict__ W2, const float* __restrict__ b2,
    const float* __restrict__ Wh, float* __restrict__ out,
    float* __restrict__ tail) {
    __shared__ __align__(16) float s_cb[DICT * EMB];   // codebook
    __shared__ float s_cn[DICT];                       // ||c||^2
    __shared__ float s_w1[EMB * HID];
    __shared__ float s_b1[HID];
    __shared__ float s_w2[HID * EMB];
    __shared__ float s_b2[EMB];
    __shared__ float s_wh[EMB * POUT];                 // W_hyper (32 KB)
    __shared__ float s_usage[DICT];
    __shared__ float s_loss;

    const int tid = threadIdx.x;
    for (int i = tid; i < DICT * EMB; i += TPB) s_cb[i] = codebook[i];
    for (int i = tid; i < EMB * HID; i += TPB) s_w1[i] = W1[i];
    for (int i = tid; i < HID * EMB; i += TPB) s_w2[i] = W2[i];
    for (int i = tid; i < EMB * POUT; i += TPB) s_wh[i] = Wh[i];
    if (tid < HID) s_b1[tid] = b1[tid];
    if (tid < EMB) s_b2[tid] = b2[tid];
    if (tid < DICT) s_usage[tid] = 0.0f;
    if (tid == 0) s_loss = 0.0f;
    __syncthreads();
    if (tid < DICT) {
        float s = 0.0f;
#pragma unroll
        for (int k = 0; k < EMB; k++) { float v = s_cb[tid * EMB + k]; s = fmaf(v, v, s); }
        s_cn[tid] = s;
    }
    __syncthreads();

    const int lane  = tid & 31;
    const int wave  = tid >> 5;
    const bool hi   = lane >= 16;        // upper half-wave of the lane pair
    const int col16 = lane & 15;         // row-in-tile / N-in-tile index
    const int gwave = blockIdx.x * WPB + wave;

    for (int tile = gwave; tile < NTILES; tile += TOTAL_WAVES) {
        const int rowbase = tile << 4;
        const int row = rowbase + col16;

        // ---- load this lane's embedding row (both halves hold the same row) ----
        const float4* er = (const float4*)(embed + (size_t)row * EMB);
        float4 e0 = er[0], e1 = er[1], e2 = er[2], e3 = er[3];

        // ---- nearest codebook entry: halves split even/odd entries ----
        float best = 3.4e38f;
        int bidx = 0;
        for (int c = hi ? 1 : 0; c < DICT; c += 2) {
            const float4* cr = (const float4*)(s_cb + c * EMB);
            float dot = dot4(e0, cr[0]) + dot4(e1, cr[1]) + dot4(e2, cr[2]) + dot4(e3, cr[3]);
            float d = fmaf(-2.0f, dot, s_cn[c]);
            if (d < best) { best = d; bidx = c; }
        }
        float ob = __shfl_xor(best, 16, 32);
        int   oi = __shfl_xor(bidx, 16, 32);
        if (ob < best || (ob == best && oi < bidx)) { best = ob; bidx = oi; }

        // ---- gather quantized vector ----
        const float4* qr = (const float4*)(s_cb + bidx * EMB);
        float4 q0 = qr[0], q1 = qr[1], q2 = qr[2], q3 = qr[3];

        // ---- loss partial + usage (low half lanes carry the 16 unique rows) ----
        float dsum = dsq4(q0, e0) + dsq4(q1, e1) + dsq4(q2, e2) + dsq4(q3, e3);
        dsum += __shfl_xor(dsum, 1, 32);
        dsum += __shfl_xor(dsum, 2, 32);
        dsum += __shfl_xor(dsum, 4, 32);
        dsum += __shfl_xor(dsum, 8, 32);
        if (lane == 0) atomicAdd(&s_loss, dsum);
        if (!hi) atomicAdd(&s_usage[bidx], 1.0f);

        // ---- MLP: 16 -> 32 -> 16, hidden dim split across the lane pair ----
        float qv[16] = {q0.x, q0.y, q0.z, q0.w, q1.x, q1.y, q1.z, q1.w,
                        q2.x, q2.y, q2.z, q2.w, q3.x, q3.y, q3.z, q3.w};
        const int joff = hi ? 16 : 0;
        float h1[16];
#pragma unroll
        for (int j = 0; j < 16; j++) {
            float s = s_b1[j + joff];
#pragma unroll
            for (int k = 0; k < 16; k++) s = fmaf(qv[k], s_w1[k * HID + j + joff], s);
            h1[j] = lrelu(s);
        }
        float h2[16];
#pragma unroll
        for (int n = 0; n < 16; n++) {
            float s = 0.0f;
#pragma unroll
            for (int j = 0; j < 16; j++) s = fmaf(h1[j], s_w2[(j + joff) * EMB + n], s);
            s += __shfl_xor(s, 16, 32);           // combine pair partials
            s += s_b2[n];
            h2[n] = lrelu(s);                      // full h2 replicated in both halves
        }

        // ---- hypernet matmul: h2[16x16 tile] @ W_hyper[16x512] via f32 WMMA ----
        // A 16x4 layout: lanes 0-15 -> K=0,1 ; lanes 16-31 -> K=2,3 (per K-slice)
        float* obase = out + (size_t)rowbase * POUT + (hi ? (size_t)8 * POUT : 0) + col16;
        const int rbase = hi ? 2 : 0;
        for (int nt = 0; nt < 32; ++nt) {
            v8f acc = {};
#pragma unroll
            for (int kc = 0; kc < 4; kc++) {
                v2f a, b;
                a.x = hi ? h2[kc * 4 + 2] : h2[kc * 4 + 0];
                a.y = hi ? h2[kc * 4 + 3] : h2[kc * 4 + 1];
                const int rr = kc * 4 + rbase;
                b.x = s_wh[rr * POUT + nt * 16 + col16];
                b.y = s_wh[(rr + 1) * POUT + nt * 16 + col16];
                acc = __builtin_amdgcn_wmma_f32_16x16x4_f32(
                    false, a, false, b, (short)0, acc, false, false);
            }
            float* op = obase + nt * 16;
#pragma unroll
            for (int r = 0; r < 8; r++) op[(size_t)r * POUT] = acc[r];
        }
    }

    // ---- flush block-local usage / loss ----
    __syncthreads();
    if (tid < DICT) {
        float u = s_usage[tid];
        if (u != 0.0f) atomicAdd(&tail[1 + tid], u);
    }
    if (tid == 0) atomicAdd(&tail[0], s_loss * (1.25f / (float)(B_ROWS * EMB)));
}

extern "C" void kernel_launch(void* const* d_in, const int* in_sizes, int n_in,
                              void* d_out, int out_size, void* d_ws, size_t ws_size,
                              hipStream_t stream) {
    (void)in_sizes; (void)n_in; (void)out_size; (void)d_ws; (void)ws_size;
    const float* embed    = (const float*)d_in[0];
    const float* codebook = (const float*)d_in[1];
    const float* W1       = (const float*)d_in[2];
    const float* b1       = (const float*)d_in[3];
    const float* W2       = (const float*)d_in[4];
    const float* b2       = (const float*)d_in[5];
    const float* Wh       = (const float*)d_in[6];
    const float* usage    = (const float*)d_in[7];
    float* out  = (float*)d_out;
    float* tail = out + (size_t)B_ROWS * POUT;   // [loss, usage_new[100]]

    init_tail_kernel<<<1, 128, 0, stream>>>(usage, tail);
    vqvae_hyper_kernel<<<NBLOCKS, TPB, 0, stream>>>(embed, codebook, W1, b1, W2,
                                                    b2, Wh, out, tail);
}